// GraphNN_10685878632725
// MI455X (gfx1250) — compile-verified
//
#include <hip/hip_runtime.h>
#include <math.h>

typedef __attribute__((ext_vector_type(2))) float v2f;
typedef __attribute__((ext_vector_type(8))) float v8f;

#define BN_EPS 1e-5f

// ---------------------------------------------------------------------------
// WMMA fp32 GEMM:  C[M,NC] = A[M,K] @ B[K,NC], each output row scaled by
// dinv[row] (folds the d^{-1/2}[src] factor of gcn_norm into the GEMM).
// One wave (32 lanes) per 16x16 output tile, 8 waves per block.
// ---------------------------------------------------------------------------
template<int K, int NC>
__global__ __launch_bounds__(256)
void gemm_wmma_scaled(const float* __restrict__ A, const float* __restrict__ B,
                      const float* __restrict__ dinv, float* __restrict__ C, int M)
{
    constexpr int TN = NC / 16;
    const int wave = threadIdx.x >> 5;
    const int lane = threadIdx.x & 31;
    const int gt = blockIdx.x * 8 + wave;
    const int tm = gt / TN;
    const int tn = gt % TN;
    const int row0 = tm * 16;
    if (row0 >= M) return;                 // wave-uniform exit: EXEC stays full

    const int lo = lane & 15;
    const int hi = lane >> 4;

    v8f acc = {};
    const float* arow = A + (size_t)(row0 + lo) * K;   // A row for this lane
    const float* bcol = B + tn * 16 + lo;              // B column for this lane

    #pragma unroll
    for (int k0 = 0; k0 < K; k0 += 4) {
        const int ka = k0 + 2 * hi;                    // lanes 0-15: K=k0,k0+1 ; 16-31: K=k0+2,k0+3
        v2f a = *(const v2f*)(arow + ka);              // 8B aligned (ka even, K even)
        v2f b;
        b.x = bcol[(size_t)ka * NC];
        b.y = bcol[(size_t)(ka + 1) * NC];
        acc = __builtin_amdgcn_wmma_f32_16x16x4_f32(
            /*neg_a=*/false, a, /*neg_b=*/false, b,
            /*c_mod=*/(short)0, acc, /*reuse_a=*/false, /*reuse_b=*/false);
    }

    #pragma unroll
    for (int r = 0; r < 8; ++r) {
        const int row = row0 + 8 * hi + r;             // C/D layout: VGPR r -> M = r + 8*hi
        C[(size_t)row * NC + tn * 16 + lo] = acc[r] * dinv[row];
    }
}

// ---------------------------------------------------------------------------
// Small elementwise kernels
// ---------------------------------------------------------------------------
__global__ void fill_f32(float* __restrict__ p, float v, int n)
{
    int i = blockIdx.x * blockDim.x + threadIdx.x;
    if (i < n) p[i] = v;
}

__global__ void deg_accum(const int* __restrict__ dsts, float* __restrict__ deg, int E)
{
    int e = blockIdx.x * blockDim.x + threadIdx.x;
    if (e < E) atomicAdd(&deg[dsts[e]], 1.0f);
}

__global__ void rsqrt_k(const float* __restrict__ deg, float* __restrict__ dinv, int n)
{
    int i = blockIdx.x * blockDim.x + threadIdx.x;
    if (i < n) dinv[i] = rsqrtf(deg[i]);       // deg >= 1 always (self loop)
}

// agg[i*F+f] = hs[i*F+f]*dinv[i] + bias[f]   (self-loop message + bias; hs is
// already pre-scaled by dinv[i], so hs*dinv = h * d^{-1} = self-loop norm)
template<int F>
__global__ void selfloop_bias(const float* __restrict__ hs, const float* __restrict__ dinv,
                              const float* __restrict__ bias, float* __restrict__ agg, int total)
{
    int t = blockIdx.x * blockDim.x + threadIdx.x;
    if (t >= total) return;
    constexpr int SH = (F == 128) ? 7 : 6;
    int i = t >> SH;
    int f = t & (F - 1);
    agg[t] = hs[t] * dinv[i] + bias[f];
}

// ---------------------------------------------------------------------------
// Edge scatter: one thread per edge; gathers the pre-scaled source row from
// L2-resident hs, scales by dinv[dst], atomically accumulates into agg[dst].
// ---------------------------------------------------------------------------
template<int F>
__global__ __launch_bounds__(256)
void scatter_edges(const float* __restrict__ hs, const float* __restrict__ dinv,
                   const int* __restrict__ srcs, const int* __restrict__ dsts,
                   float* __restrict__ agg, int E)
{
    int e = blockIdx.x * blockDim.x + threadIdx.x;
    if (e >= E) return;
    const int r = srcs[e];
    const int c = dsts[e];
    const float s = dinv[c];
    const float4* hrow = (const float4*)(hs + (size_t)r * F);
    float* out = agg + (size_t)c * F;
    #pragma unroll 8
    for (int q = 0; q < F / 4; ++q) {
        float4 v = hrow[q];
        atomicAdd(out + 4 * q + 0, v.x * s);
        atomicAdd(out + 4 * q + 1, v.y * s);
        atomicAdd(out + 4 * q + 2, v.z * s);
        atomicAdd(out + 4 * q + 3, v.w * s);
    }
}

// ---------------------------------------------------------------------------
// BatchNorm1d (training-mode batch stats, biased variance) over [N,64]
// ---------------------------------------------------------------------------
__global__ __launch_bounds__(256)
void bn_stats(const float* __restrict__ x, float* __restrict__ sums, int Nrows)
{
    __shared__ float ssum[64];
    __shared__ float ssq[64];
    if (threadIdx.x < 64) { ssum[threadIdx.x] = 0.f; ssq[threadIdx.x] = 0.f; }
    __syncthreads();
    const int f = threadIdx.x & 63;
    int row = blockIdx.x * 4 + (threadIdx.x >> 6);
    const int stride = gridDim.x * 4;
    float s = 0.f, q = 0.f;
    for (; row < Nrows; row += stride) {
        float v = x[(size_t)row * 64 + f];
        s += v; q += v * v;
    }
    atomicAdd(&ssum[f], s);
    atomicAdd(&ssq[f], q);
    __syncthreads();
    if (threadIdx.x < 64) {
        atomicAdd(&sums[threadIdx.x], ssum[threadIdx.x]);
        atomicAdd(&sums[64 + threadIdx.x], ssq[threadIdx.x]);
    }
}

__global__ void bn_apply(float* __restrict__ x, const float* __restrict__ sums,
                         const float* __restrict__ gamma, const float* __restrict__ beta,
                         int Nrows)
{
    int t = blockIdx.x * blockDim.x + threadIdx.x;
    int total = Nrows * 64;
    if (t >= total) return;
    int f = t & 63;
    float invN = 1.0f / (float)Nrows;
    float mean = sums[f] * invN;
    float var  = sums[64 + f] * invN - mean * mean;
    float inv  = rsqrtf(var + BN_EPS);
    x[t] = (x[t] - mean) * inv * gamma[f] + beta[f];
}

// ---------------------------------------------------------------------------
// Launcher
// ---------------------------------------------------------------------------
extern "C" void kernel_launch(void* const* d_in, const int* in_sizes, int n_in,
                              void* d_out, int out_size, void* d_ws, size_t ws_size,
                              hipStream_t stream)
{
    const float* emb   = (const float*)d_in[0];   // [N,64]
    const float* W1    = (const float*)d_in[1];   // [64,128]
    const float* b1    = (const float*)d_in[2];   // [128]
    const float* W2    = (const float*)d_in[3];   // [128,64]
    const float* b2    = (const float*)d_in[4];   // [64]
    const float* gamma = (const float*)d_in[5];   // [64]
    const float* beta  = (const float*)d_in[6];   // [64]
    const int*   eidx  = (const int*)d_in[7];     // [2,E]

    const int N = in_sizes[0] / 64;
    const int E = in_sizes[7] / 2;
    const int* srcs = eidx;          // edge_index[0] (source j)
    const int* dsts = eidx + E;      // edge_index[1] (target i)

    float* ws   = (float*)d_ws;
    float* deg  = ws;                                // N
    float* dinv = ws + N;                            // N
    float* buf1 = ws + 2 * (size_t)N;                // N*128  (h scaled / h2 scaled)
    float* buf2 = buf1 + (size_t)N * 128;            // N*128  (agg1)
    float* sums = buf2 + (size_t)N * 128;            // 128
    float* out  = (float*)d_out;                     // N*64   (agg2 -> BN in-place)

    const int threads = 256;

    // 1) degrees (self-loop folded into init value 1.0)
    fill_f32<<<(N + threads - 1) / threads, threads, 0, stream>>>(deg, 1.0f, N);
    deg_accum<<<(E + threads - 1) / threads, threads, 0, stream>>>(dsts, deg, E);
    rsqrt_k<<<(N + threads - 1) / threads, threads, 0, stream>>>(deg, dinv, N);

    // 2) layer 1: h_scaled = (emb @ W1) * dinv[row]
    {
        const int tiles = ((N + 15) / 16) * (128 / 16);
        gemm_wmma_scaled<64, 128><<<(tiles + 7) / 8, threads, 0, stream>>>(emb, W1, dinv, buf1, N);
    }
    selfloop_bias<128><<<((size_t)N * 128 + threads - 1) / threads, threads, 0, stream>>>(
        buf1, dinv, b1, buf2, N * 128);
    scatter_edges<128><<<(E + threads - 1) / threads, threads, 0, stream>>>(
        buf1, dinv, srcs, dsts, buf2, E);

    // 3) layer 2: h2_scaled = (agg1 @ W2) * dinv[row]
    {
        const int tiles = ((N + 15) / 16) * (64 / 16);
        gemm_wmma_scaled<128, 64><<<(tiles + 7) / 8, threads, 0, stream>>>(buf2, W2, dinv, buf1, N);
    }
    selfloop_bias<64><<<((size_t)N * 64 + threads - 1) / threads, threads, 0, stream>>>(
        buf1, dinv, b2, out, N * 64);
    scatter_edges<64><<<(E + threads - 1) / threads, threads, 0, stream>>>(
        buf1, dinv, srcs, dsts, out, E);

    // 4) BatchNorm1d (batch stats), in-place on d_out
    fill_f32<<<1, 128, 0, stream>>>(sums, 0.0f, 128);
    bn_stats<<<512, threads, 0, stream>>>(out, sums, N);
    bn_apply<<<((size_t)N * 64 + threads - 1) / threads, threads, 0, stream>>>(
        out, sums, gamma, beta, N);
}